// DynamicGeometricRotation_64922725646672
// MI455X (gfx1250) — compile-verified
//
#include <hip/hip_runtime.h>
#include <hip/hip_bf16.h>

typedef __attribute__((ext_vector_type(2))) float v2f;
typedef __attribute__((ext_vector_type(8))) float v8f;

#define B_    16
#define S_    8192
#define D_    128
#define H_    512
#define NROT  3
#define LDA   132      // padded LDS row stride (floats) to avoid bank conflicts
#define NTERM 12       // Taylor terms for expm (||G|| ~ 0.4 -> fp32-exact)

// ---------------------------------------------------------------------------
// WMMA helper: D = A(16x4 f32) * B(4x16 f32) + C, wave32.
// ---------------------------------------------------------------------------
__device__ inline v8f wmma4(v2f a, v2f b, v8f c) {
    return __builtin_amdgcn_wmma_f32_16x16x4_f32(
        /*neg_a=*/false, a, /*neg_b=*/false, b,
        /*c_mod=*/(short)0, c, /*reuse_a=*/false, /*reuse_b=*/false);
}

// 16x16 output tile of a 128-deep matmul, A and B row-major in LDS (ld = LDA).
__device__ inline v8f tile_mm128(const float* sA, const float* sB,
                                 int mBase, int nBase, int half, int lr) {
    v8f c;
    #pragma unroll
    for (int r = 0; r < 8; ++r) c[r] = 0.f;
    #pragma unroll 8
    for (int kk = 0; kk < 128; kk += 4) {
        int k0 = kk + half * 2;
        v2f a, b;
        a[0] = sA[(mBase + lr) * LDA + k0];
        a[1] = sA[(mBase + lr) * LDA + k0 + 1];
        b[0] = sB[k0 * LDA + nBase + lr];
        b[1] = sB[(k0 + 1) * LDA + nBase + lr];
        c = wmma4(a, b, c);
    }
    return c;
}

// ---------------------------------------------------------------------------
// gfx1250 async global -> LDS copy (no VGPR staging; tracked by ASYNCcnt).
// lds_off: byte offset within the wave's LDS allocation (low 32 bits of a
// flat LDS pointer, per the ISA's flat->LDS truncation rule).
// ---------------------------------------------------------------------------
__device__ inline void async_ld_b128(unsigned lds_off, const void* gaddr) {
    asm volatile("global_load_async_to_lds_b128 %0, %1, off"
                 :: "v"(lds_off), "v"(gaddr)
                 : "memory");
}
__device__ inline void wait_asynccnt0() {
    asm volatile("s_wait_asynccnt 0x0" ::: "memory");
}

// ---------------------------------------------------------------------------
// 1) zero the pooled accumulator
// ---------------------------------------------------------------------------
__global__ __launch_bounds__(256) void k_zero(float* p, int n) {
    int i = blockIdx.x * blockDim.x + threadIdx.x;
    if (i < n) p[i] = 0.f;
}

// ---------------------------------------------------------------------------
// 2) pooled[b,d] = mean_s x[b,s,d]  (split-S partials + f32 atomics)
// ---------------------------------------------------------------------------
__global__ __launch_bounds__(128) void k_pool(const float* __restrict__ x,
                                              float* __restrict__ pooled) {
    int b = blockIdx.x >> 6;
    int s0 = (blockIdx.x & 63) * 128;
    int d = threadIdx.x;
    const float* px = x + (size_t)b * S_ * D_ + (size_t)s0 * D_ + d;
    float sum = 0.f;
    #pragma unroll 8
    for (int s = 0; s < 128; ++s) sum += px[(size_t)s * D_];
    atomicAdd(&pooled[b * D_ + d], sum * (1.0f / (float)S_));
}

// ---------------------------------------------------------------------------
// 3) h[b,j] = gelu_exact(pooled[b,:] . W1[j,:] + b1[j])
// ---------------------------------------------------------------------------
__global__ __launch_bounds__(256) void k_hidden(const float* __restrict__ pooled,
                                                const float* __restrict__ W1,
                                                const float* __restrict__ b1,
                                                float* __restrict__ h) {
    int idx = blockIdx.x * blockDim.x + threadIdx.x;   // B_*H_ threads
    int b = idx >> 9;
    int j = idx & 511;
    const float* p = pooled + b * D_;
    const float* w = W1 + j * D_;
    float acc = b1[j];
    #pragma unroll 8
    for (int k = 0; k < D_; ++k) acc += p[k] * w[k];
    h[idx] = 0.5f * acc * (1.0f + erff(acc * 0.70710678118654752f));
}

// ---------------------------------------------------------------------------
// 4) P = h @ W2^T + b2   (M=16, K=512, N=49152) via WMMA f32 16x16x4.
// ---------------------------------------------------------------------------
__global__ __launch_bounds__(256) void k_params(const float* __restrict__ h,
                                                const float* __restrict__ W2,
                                                const float* __restrict__ b2,
                                                float* __restrict__ P) {
    extern __shared__ float sH[];          // 16 x 516 floats
    const int LDH = 516;
    int tid = threadIdx.x;
    for (int i = tid; i < B_ * H_; i += 256)
        sH[(i >> 9) * LDH + (i & 511)] = h[i];
    __syncthreads();

    int wave = tid >> 5, lane = tid & 31;
    int half = lane >> 4, lr = lane & 15;
    int nBase = (blockIdx.x * 8 + wave) * 16;
    const float* wrow = W2 + (size_t)(nBase + lr) * H_;

    v8f acc;
    #pragma unroll
    for (int r = 0; r < 8; ++r) acc[r] = 0.f;
    #pragma unroll 4
    for (int kk = 0; kk < H_; kk += 4) {
        int k0 = kk + half * 2;
        v2f a, b;
        a[0] = sH[lr * LDH + k0];
        a[1] = sH[lr * LDH + k0 + 1];
        b[0] = wrow[k0];
        b[1] = wrow[k0 + 1];
        acc = wmma4(a, b, acc);
    }
    float bias = b2[nBase + lr];
    #pragma unroll
    for (int r = 0; r < 8; ++r) {
        int m = half * 8 + r;              // batch index (M dimension)
        P[(size_t)m * (NROT * D_ * D_) + nBase + lr] = acc[r] + bias;
    }
}

// ---------------------------------------------------------------------------
// 5) G = 0.5 * (P - P^T) per (b, rot) 128x128 matrix
// ---------------------------------------------------------------------------
__global__ __launch_bounds__(256) void k_skew(const float* __restrict__ P,
                                              float* __restrict__ G) {
    int idx = blockIdx.x * blockDim.x + threadIdx.x;   // B*NROT*128*128
    int rc  = idx & (D_ * D_ - 1);
    int mat = idx >> 14;
    int r = rc >> 7, c = rc & 127;
    const float* p = P + (size_t)mat * D_ * D_;
    G[idx] = 0.5f * (p[r * D_ + c] - p[c * D_ + r]);
}

// ---------------------------------------------------------------------------
// 6) Rot = expm(G) via Taylor, one block (8 waves) per 128x128 matrix.
// ---------------------------------------------------------------------------
__global__ __launch_bounds__(256) void k_expm(const float* __restrict__ G,
                                              float* __restrict__ Rot) {
    extern __shared__ float smem[];
    float* sG  = smem;                 // 128*LDA
    float* sT0 = smem + 128 * LDA;
    float* sT1 = smem + 2 * 128 * LDA;
    int tid = threadIdx.x, wave = tid >> 5, lane = tid & 31;
    int half = lane >> 4, lr = lane & 15;

    const float* g = G + (size_t)blockIdx.x * D_ * D_;
    for (int i = tid; i < D_ * D_; i += 256) {
        int r = i >> 7, c = i & 127;
        float v = g[i];
        sG[r * LDA + c]  = v;
        sT0[r * LDA + c] = v;
    }
    __syncthreads();

    int mBase = wave * 16;
    v8f acc[8];
    #pragma unroll
    for (int nt = 0; nt < 8; ++nt)
        for (int r = 0; r < 8; ++r) {
            int row = mBase + half * 8 + r, col = nt * 16 + lr;
            acc[nt][r] = sG[row * LDA + col] + (row == col ? 1.f : 0.f);
        }

    float* src = sT0;
    float* dst = sT1;
    for (int k = 2; k <= NTERM; ++k) {
        float invk = 1.f / (float)k;
        for (int nt = 0; nt < 8; ++nt) {
            v8f c = tile_mm128(src, sG, mBase, nt * 16, half, lr);
            #pragma unroll
            for (int r = 0; r < 8; ++r) {
                c[r] *= invk;
                acc[nt][r] += c[r];
                dst[(mBase + half * 8 + r) * LDA + nt * 16 + lr] = c[r];
            }
        }
        __syncthreads();
        float* t = src; src = dst; dst = t;
    }

    float* out = Rot + (size_t)blockIdx.x * D_ * D_;
    #pragma unroll
    for (int nt = 0; nt < 8; ++nt)
        for (int r = 0; r < 8; ++r)
            out[(mBase + half * 8 + r) * D_ + nt * 16 + lr] = acc[nt][r];
}

// ---------------------------------------------------------------------------
// 7) R_b = rot0 @ rot1 @ rot2  (one block per batch)
// ---------------------------------------------------------------------------
__global__ __launch_bounds__(256) void k_combine(const float* __restrict__ Rot,
                                                 float* __restrict__ R) {
    extern __shared__ float smem[];
    float* sA = smem;                  // 128*LDA
    float* sB = smem + 128 * LDA;
    int tid = threadIdx.x, wave = tid >> 5, lane = tid & 31;
    int half = lane >> 4, lr = lane & 15;
    int b = blockIdx.x;
    const float* r0 = Rot + ((size_t)b * NROT + 0) * D_ * D_;
    const float* r1 = Rot + ((size_t)b * NROT + 1) * D_ * D_;
    const float* r2 = Rot + ((size_t)b * NROT + 2) * D_ * D_;

    for (int i = tid; i < D_ * D_; i += 256) {
        int r = i >> 7, c = i & 127;
        sA[r * LDA + c] = r0[i];
        sB[r * LDA + c] = r1[i];
    }
    __syncthreads();

    int mBase = wave * 16;
    v8f t[8];
    for (int nt = 0; nt < 8; ++nt)
        t[nt] = tile_mm128(sA, sB, mBase, nt * 16, half, lr);
    __syncthreads();                   // all waves done reading sA/sB

    // tmp -> sA (each wave writes only its own rows), rot2 -> sB
    for (int nt = 0; nt < 8; ++nt)
        for (int r = 0; r < 8; ++r)
            sA[(mBase + half * 8 + r) * LDA + nt * 16 + lr] = t[nt][r];
    __syncthreads();
    for (int i = tid; i < D_ * D_; i += 256) {
        int r = i >> 7, c = i & 127;
        sB[r * LDA + c] = r2[i];
    }
    __syncthreads();

    float* out = R + (size_t)b * D_ * D_;
    for (int nt = 0; nt < 8; ++nt) {
        v8f c = tile_mm128(sA, sB, mBase, nt * 16, half, lr);
        #pragma unroll
        for (int r = 0; r < 8; ++r)
            out[(mBase + half * 8 + r) * D_ + nt * 16 + lr] = c[r];
    }
}

// ---------------------------------------------------------------------------
// 8) out[b, s0:s0+128, :] = x[b, s0:s0+128, :] @ R_b   (HBM-bound pass)
//    X tile and R staged into LDS via gfx1250 async global->LDS B128 copies
//    (no VGPR round-trip), then fp32 WMMA from LDS, direct global stores.
// ---------------------------------------------------------------------------
__global__ __launch_bounds__(256) void k_apply(const float* __restrict__ x,
                                               const float* __restrict__ R,
                                               float* __restrict__ out) {
    extern __shared__ float smem[];
    float* sX = smem;                  // 128*LDA
    float* sR = smem + 128 * LDA;
    int tid = threadIdx.x, wave = tid >> 5, lane = tid & 31;
    int half = lane >> 4, lr = lane & 15;
    int b = blockIdx.x >> 6;
    int s0 = (blockIdx.x & 63) * 128;

    const float4* px4 = (const float4*)(x + (size_t)b * S_ * D_ + (size_t)s0 * D_);
    const float4* pr4 = (const float4*)(R + (size_t)b * D_ * D_);
    for (int i = tid; i < D_ * D_ / 4; i += 256) {   // 16B chunks, coalesced
        int r = i >> 5, c4 = (i & 31) * 4;
        unsigned dx = (unsigned)(uintptr_t)(&sX[r * LDA + c4]);
        unsigned dr = (unsigned)(uintptr_t)(&sR[r * LDA + c4]);
        async_ld_b128(dx, &px4[i]);
        async_ld_b128(dr, &pr4[i]);
    }
    wait_asynccnt0();                  // this wave's async fills landed in LDS
    __syncthreads();                   // ... and everyone else's too

    int mBase = wave * 16;
    float* po = out + (size_t)b * S_ * D_ + (size_t)s0 * D_;
    for (int nt = 0; nt < 8; ++nt) {
        v8f c = tile_mm128(sX, sR, mBase, nt * 16, half, lr);
        #pragma unroll
        for (int r = 0; r < 8; ++r)
            po[(mBase + half * 8 + r) * D_ + nt * 16 + lr] = c[r];
    }
}

// ---------------------------------------------------------------------------
extern "C" void kernel_launch(void* const* d_in, const int* in_sizes, int n_in,
                              void* d_out, int out_size, void* d_ws, size_t ws_size,
                              hipStream_t stream) {
    const float* x  = (const float*)d_in[0];
    const float* W1 = (const float*)d_in[1];
    const float* b1 = (const float*)d_in[2];
    const float* W2 = (const float*)d_in[3];
    const float* b2 = (const float*)d_in[4];
    float* out = (float*)d_out;

    float* ws     = (float*)d_ws;
    float* pooled = ws;                                   // 2048
    float* h      = pooled + B_ * D_;                     // 8192
    float* P      = h + B_ * H_;                          // 786432
    float* G      = P + (size_t)B_ * NROT * D_ * D_;      // 786432
    float* Rot    = G + (size_t)B_ * NROT * D_ * D_;      // 786432
    float* Rm     = Rot + (size_t)B_ * NROT * D_ * D_;    // 262144 (~10.5 MB total)

    const size_t lds_mat = (size_t)128 * LDA * sizeof(float);

    k_zero<<<(B_ * D_ + 255) / 256, 256, 0, stream>>>(pooled, B_ * D_);
    k_pool<<<B_ * (S_ / 128), 128, 0, stream>>>(x, pooled);
    k_hidden<<<(B_ * H_) / 256, 256, 0, stream>>>(pooled, W1, b1, h);
    k_params<<<(NROT * D_ * D_ / 16) / 8, 256, 16 * 516 * sizeof(float), stream>>>(h, W2, b2, P);
    k_skew<<<(B_ * NROT * D_ * D_) / 256, 256, 0, stream>>>(P, G);
    k_expm<<<B_ * NROT, 256, 3 * lds_mat, stream>>>(G, Rot);
    k_combine<<<B_, 256, 2 * lds_mat, stream>>>(Rot, Rm);
    k_apply<<<B_ * (S_ / 128), 256, 2 * lds_mat, stream>>>(x, Rm, out);
}